// HexPool_68805376082188
// MI455X (gfx1250) — compile-verified
//
#include <hip/hip_runtime.h>
#include <hip/hip_bf16.h>

// HexPool: out[i, :] = max over k in 0..6 of x[neigh_indices[i,k], :]
//   x:             (655362, 64) f32   (only first 163842 rows are gathered)
//   neigh_indices: (163842, 7)  i64   (values in [0, 163842))
//   out:           (163842, 64) f32
//
// Pure gather + max-pool: memory-bound (~93 MB HBM traffic -> ~4 us floor at
// 23.3 TB/s). The 42 MB gather source fits in the 192 MB L2, so the 7x gather
// reuse is served from L2. No matmul structure -> WMMA inapplicable; optimal
// instruction mix is global_load_b128 gathers + v_max_num_f32 + NT stores.

#define HP_N_OUT 163842
#define HP_D     64
#define HP_K     7

typedef __attribute__((ext_vector_type(4))) float f32x4;

__device__ __forceinline__ f32x4 vmax4(f32x4 a, f32x4 b) {
    f32x4 r;
    r.x = __builtin_fmaxf(a.x, b.x);
    r.y = __builtin_fmaxf(a.y, b.y);
    r.z = __builtin_fmaxf(a.z, b.z);
    r.w = __builtin_fmaxf(a.w, b.w);
    return r;
}

__global__ __launch_bounds__(256) void HexPool_68805376082188_kernel(
    const float* __restrict__ x,
    const long long* __restrict__ nidx,
    float* __restrict__ out)
{
    // 16 lanes per output row, each lane owns a float4 (16B) of the 256B row.
    // A wave32 covers 2 rows with fully coalesced b128 traffic.
    const int tasks  = HP_N_OUT * 16;
    const int stride = (int)(gridDim.x * blockDim.x);

    for (int gid = (int)(blockIdx.x * blockDim.x + threadIdx.x);
         gid < tasks; gid += stride)
    {
        const int row = gid >> 4;          // output row
        const int col = (gid & 15) << 2;   // float offset within row (0..60)

        // Prefetch next iteration's index line (gfx1250: global_prefetch_b8).
        const int nrow = (gid + stride) >> 4;
        if (nrow < HP_N_OUT)
            __builtin_prefetch((const void*)(nidx + (size_t)nrow * HP_K), 0, 3);

        const long long* ip = nidx + (size_t)row * HP_K;
        // 16 lanes of this row group request the same cachelines -> merged by
        // the wave coalescer; plain loads keep them L2-resident (streamed once).
        const long long i0 = ip[0];
        const long long i1 = ip[1];
        const long long i2 = ip[2];
        const long long i3 = ip[3];
        const long long i4 = ip[4];
        const long long i5 = ip[5];
        const long long i6 = ip[6];

        const float* xb = x + col;
        // Issue all 7 b128 gathers before reducing: 7 loads in flight per lane.
        const f32x4 v0 = *(const f32x4*)(xb + (size_t)i0 * HP_D);
        const f32x4 v1 = *(const f32x4*)(xb + (size_t)i1 * HP_D);
        const f32x4 v2 = *(const f32x4*)(xb + (size_t)i2 * HP_D);
        const f32x4 v3 = *(const f32x4*)(xb + (size_t)i3 * HP_D);
        const f32x4 v4 = *(const f32x4*)(xb + (size_t)i4 * HP_D);
        const f32x4 v5 = *(const f32x4*)(xb + (size_t)i5 * HP_D);
        const f32x4 v6 = *(const f32x4*)(xb + (size_t)i6 * HP_D);

        // Max tree (depth 3) -> 24 v_max_num_f32 per lane-task.
        const f32x4 m01   = vmax4(v0, v1);
        const f32x4 m23   = vmax4(v2, v3);
        const f32x4 m45   = vmax4(v4, v5);
        const f32x4 m0123 = vmax4(m01, m23);
        const f32x4 m456  = vmax4(m45, v6);
        const f32x4 m     = vmax4(m0123, m456);

        // Streamed output: non-temporal store so the 42 MB result stream does
        // not evict the L2-resident gather source.
        __builtin_nontemporal_store(m, (f32x4*)(out + (size_t)row * HP_D + col));
    }
}

extern "C" void kernel_launch(void* const* d_in, const int* in_sizes, int n_in,
                              void* d_out, int out_size, void* d_ws, size_t ws_size,
                              hipStream_t stream) {
    (void)in_sizes; (void)n_in; (void)out_size; (void)d_ws; (void)ws_size;

    const float*     x    = (const float*)d_in[0];       // (655362, 64) f32
    const long long* nidx = (const long long*)d_in[1];   // (163842, 7)  i64
    float*           out  = (float*)d_out;               // (163842, 64) f32

    // Grid-stride persistent-ish launch: 2048 blocks x 256 threads (8 waves
    // per block under wave32) -> ~5 iterations per thread; enough waves to
    // saturate every WGP while keeping the index prefetch useful.
    const dim3 grid(2048);
    const dim3 block(256);
    HexPool_68805376082188_kernel<<<grid, block, 0, stream>>>(x, nidx, out);
}